// SS2D_27075473834178
// MI455X (gfx1250) — compile-verified
//
#include <hip/hip_runtime.h>
#include <hip/hip_bf16.h>

// ---------------------------------------------------------------------------
// SS2D (vision Mamba) for gfx1250. All dense stages use V_WMMA_F32_16X16X4_F32
// (native fp32 WMMA, exact fp32 semantics). Scans are parallel over
// rows x channels with the state held in registers. Elementwise activations
// use v_rcp_f32 instead of IEEE divide; scan streams use b128 loads and
// global_prefetch_b8 for the next step.
// ---------------------------------------------------------------------------

#define C_INc    20
#define D_INNERc 32
#define D_STATEc 8
#define OUT_CHc  20
#define DT_RANKc 4
#define Bn       4
#define Hn       384
#define Wn       384
#define NPIX     (Bn * Hn * Wn)      // 589824
#define EPSc     1e-5f
#define TILE     16

typedef __attribute__((ext_vector_type(2))) float v2f;
typedef __attribute__((ext_vector_type(8))) float v8f;

// D(16x16) += A(16xK) * W^T(Kx16), fp32 WMMA, K % 4 == 0.
// Als: LDS, row-major 16 x ldA.  Wt: global (OUTS x ldW) row-major, we use
// rows [n_base, n_base+15] as the 16 columns of B (clamped to n_max to avoid
// OOB reads for partial tiles; invalid columns are simply not stored).
// VGPR layouts per CDNA5 ISA 7.12.2:
//   A 16x4 f32:  lanes 0-15 -> M=lane, K={k0,k0+1}; lanes 16-31 -> K={k0+2,k0+3}
//   B  4x16 f32: mirrored (lanes 0-15 rows K={k0,k0+1}, N=lane&15)
//   C/D 16x16:   c[r] = D[M = r + 8*(lane>=16)][N = lane&15]
__device__ inline v8f wmma_tile_f32(const float* __restrict__ Als, int ldA,
                                    const float* __restrict__ Wt, int ldW,
                                    int K, int n_base, int n_max, int lane) {
  v8f c = {0.f, 0.f, 0.f, 0.f, 0.f, 0.f, 0.f, 0.f};
  const int row = lane & 15;
  const int hi  = (lane >> 4) & 1;
  int o = n_base + row;
  if (o > n_max) o = n_max;            // clamp (results discarded by caller)
  for (int k0 = 0; k0 < K; k0 += 4) {
    const int kk = k0 + hi * 2;
    v2f a, b;
    a[0] = Als[row * ldA + kk];
    a[1] = Als[row * ldA + kk + 1];
    b[0] = Wt[o * ldW + kk];
    b[1] = Wt[o * ldW + kk + 1];
    c = __builtin_amdgcn_wmma_f32_16x16x4_f32(false, a, false, b,
                                              (short)0, c, false, false);
  }
  return c;
}

// silu with v_rcp_f32 (1-ulp reciprocal) instead of IEEE divide
__device__ inline float silu(float v) {
  return v * __builtin_amdgcn_rcpf(1.f + __expf(-v));
}

// ---------------------------------------------------------------------------
// K1: channel-LayerNorm + in-proj (20->32) + SiLU.  One wave per 16 pixels.
// ---------------------------------------------------------------------------
__global__ __launch_bounds__(256) void k1_ln_inproj(
    const float* __restrict__ x, const float* __restrict__ ng,
    const float* __restrict__ nbias, const float* __restrict__ ipw,
    float* __restrict__ pre) {
  __shared__ float lnX[8][TILE][C_INc];
  const int wid = threadIdx.y, lane = threadIdx.x;
  const long base = ((long)blockIdx.x * 8 + wid) * TILE;
  float* Als = &lnX[wid][0][0];

  // gather 16 pixels x 20 channels from (B,C,H,W)
  for (int i = lane; i < TILE * C_INc; i += 32) {
    const int m = i / C_INc, c = i % C_INc;
    const long p = base + m;
    const long b = p / (long)(Hn * Wn);
    const long rem = p - b * (long)(Hn * Wn);
    Als[m * C_INc + c] = x[(b * C_INc + c) * (long)(Hn * Wn) + rem];
  }
  __syncthreads();

  if (lane < TILE) {                    // per-pixel LayerNorm over 20 channels
    float s = 0.f;
    for (int c = 0; c < C_INc; ++c) s += Als[lane * C_INc + c];
    const float mean = s * (1.f / C_INc);
    float v = 0.f;
    for (int c = 0; c < C_INc; ++c) {
      const float d = Als[lane * C_INc + c] - mean;
      v += d * d;
    }
    const float inv = rsqrtf(v * (1.f / C_INc) + EPSc);
    for (int c = 0; c < C_INc; ++c)
      Als[lane * C_INc + c] =
          (Als[lane * C_INc + c] - mean) * inv * ng[c] + nbias[c];
  }
  __syncthreads();

  const int row = lane & 15, hi = lane >> 4;
  for (int nt = 0; nt < 2; ++nt) {
    v8f c = wmma_tile_f32(Als, C_INc, ipw, C_INc, C_INc,
                          nt * 16, D_INNERc - 1, lane);
    const int o = nt * 16 + row;
    for (int r = 0; r < 8; ++r) {
      const long m = base + r + hi * 8;
      pre[m * D_INNERc + o] = silu(c[r]);
    }
  }
}

// ---------------------------------------------------------------------------
// K2: depthwise 3x3 SAME conv + bias + SiLU, one thread per (pixel, channel).
// ---------------------------------------------------------------------------
__global__ __launch_bounds__(256) void k2_dwconv(
    const float* __restrict__ pre, const float* __restrict__ ww,
    const float* __restrict__ wb, float* __restrict__ hbuf) {
  const long idx = (long)blockIdx.x * 256 + threadIdx.x;
  const int d = (int)(idx & (D_INNERc - 1));
  const long p = idx >> 5;
  const long b = p / (long)(Hn * Wn);
  const int rem = (int)(p - b * (long)(Hn * Wn));
  const int yy = rem / Wn, xx = rem - (rem / Wn) * Wn;
  float acc = wb[d];
  for (int ky = 0; ky < 3; ++ky) {
    const int sy = yy + ky - 1;
    if (sy < 0 || sy >= Hn) continue;
    for (int kx = 0; kx < 3; ++kx) {
      const int sx = xx + kx - 1;
      if (sx < 0 || sx >= Wn) continue;
      acc += ww[(ky * 3 + kx) * D_INNERc + d] *
             pre[((b * Hn + sy) * (long)Wn + sx) * D_INNERc + d];
    }
  }
  hbuf[idx] = silu(acc);
}

// ---------------------------------------------------------------------------
// K3a (per direction): xdbl = h @ xw^T (32->20) then dt = softplus(dtlo@dw^T+db)
// Stores dt (pixel x 32) and B,C (pixel x 16).  Two WMMA GEMMs per tile.
// ---------------------------------------------------------------------------
__global__ __launch_bounds__(256) void k3a_proj(
    const float* __restrict__ hbuf, const float* __restrict__ xw,
    const float* __restrict__ dww, const float* __restrict__ dbb,
    float* __restrict__ dt_buf, float* __restrict__ bc_buf) {
  __shared__ float Ah[8][TILE][D_INNERc];
  __shared__ float Dl[8][TILE][DT_RANKc];
  const int wid = threadIdx.y, lane = threadIdx.x;
  const long base = ((long)blockIdx.x * 8 + wid) * TILE;
  float* Als = &Ah[wid][0][0];
  float* Dls = &Dl[wid][0][0];

  {  // stage 16x32 h tile into LDS with b128 traffic
    float4* dst = (float4*)Als;
    const float4* src = (const float4*)(hbuf + base * D_INNERc);
    for (int i = lane; i < TILE * D_INNERc / 4; i += 32) dst[i] = src[i];
  }
  __syncthreads();

  const int row = lane & 15, hi = lane >> 4;
  // GEMM1: xdbl (20 outs): [0,4)=dt_lo -> LDS, [4,20) -> B,C buffer
  for (int nt = 0; nt < 2; ++nt) {
    v8f c = wmma_tile_f32(Als, D_INNERc, xw, D_INNERc, D_INNERc,
                          nt * 16, DT_RANKc + 2 * D_STATEc - 1, lane);
    const int o = nt * 16 + row;
    for (int r = 0; r < 8; ++r) {
      const int m = r + hi * 8;
      const float v = c[r];
      if (o < DT_RANKc)
        Dls[m * DT_RANKc + o] = v;
      else if (o < DT_RANKc + 2 * D_STATEc)
        bc_buf[(base + m) * 16 + (o - DT_RANKc)] = v;
    }
  }
  __syncthreads();
  // GEMM2: dt = softplus(dt_lo @ dw^T + db), K = 4 (single wmma step)
  for (int nt = 0; nt < 2; ++nt) {
    v8f c = wmma_tile_f32(Dls, DT_RANKc, dww, DT_RANKc, DT_RANKc,
                          nt * 16, D_INNERc - 1, lane);
    const int o = nt * 16 + row;
    const float bias = dbb[o];
    for (int r = 0; r < 8; ++r) {
      const int m = r + hi * 8;
      const float v = c[r] + bias;
      const float sp = (v > 20.f) ? v : log1pf(__expf(v));
      dt_buf[(base + m) * D_INNERc + o] = sp;
    }
  }
}

// ---------------------------------------------------------------------------
// K3c: selective scan.  One thread per (sequence row, channel); 8-state SSM in
// registers; reverse handled by iteration order, vertical by stride Wn.
// B/C arrive as four b128 loads (one 64B line); next step is prefetched.
// ---------------------------------------------------------------------------
__global__ __launch_bounds__(256) void k3c_scan(
    const float* __restrict__ hbuf, const float* __restrict__ dt_buf,
    const float* __restrict__ bc_buf, const float* __restrict__ Alog,
    const float* __restrict__ Dp, float* __restrict__ ybuf,
    int vertical, int reverse) {
  const int d = threadIdx.x;
  const int r = blockIdx.x * blockDim.y + threadIdx.y;
  long base; int step, len;
  if (vertical) {
    const int b = r / Wn, w = r - (r / Wn) * Wn;
    base = (long)b * Hn * Wn + w; step = Wn; len = Hn;
  } else {
    base = (long)r * Wn; step = 1; len = Wn;
  }
  float negA[D_STATEc], hst[D_STATEc];
#pragma unroll
  for (int n = 0; n < D_STATEc; ++n) {
    negA[n] = -__expf(Alog[d * D_STATEc + n]);
    hst[n] = 0.f;
  }
  const float Dv = Dp[d];
  long idx = base + (long)(reverse ? (len - 1) : 0) * step;
  const long dstep = reverse ? -(long)step : (long)step;
  for (int i = 0; i < len; ++i) {
    // prefetch next step (at the ends it is one step past the sequence; dt_buf's may lie before ws)
    const long nidx = idx + dstep;
    __builtin_prefetch(hbuf + nidx * D_INNERc + d, 0, 1);
    __builtin_prefetch(dt_buf + nidx * D_INNERc + d, 0, 1);
    __builtin_prefetch(bc_buf + nidx * 16, 0, 1);

    const float u = hbuf[idx * D_INNERc + d];
    const float dt = dt_buf[idx * D_INNERc + d];
    const float dtu = dt * u;
    const float4* bcv = (const float4*)(bc_buf + idx * 16);  // 64B line
    const float4 b0 = bcv[0], b1 = bcv[1];
    const float4 c0 = bcv[2], c1 = bcv[3];
    const float Bv[D_STATEc] = {b0.x, b0.y, b0.z, b0.w, b1.x, b1.y, b1.z, b1.w};
    const float Cv[D_STATEc] = {c0.x, c0.y, c0.z, c0.w, c1.x, c1.y, c1.z, c1.w};
    float y = 0.f;
#pragma unroll
    for (int n = 0; n < D_STATEc; ++n) {
      const float dA = __expf(dt * negA[n]);
      const float h2 = dA * hst[n] + dtu * Bv[n];
      hst[n] = h2;
      y += h2 * Cv[n];
    }
    ybuf[idx * D_INNERc + d] = y + Dv * u;
    idx += dstep;
  }
}

// ---------------------------------------------------------------------------
// K4: concat-128 LayerNorm + merge (128->32) + out (32->20) + bias, with the
// (B,H,W,20) -> (B,20,H,W) transpose fused into the store.  WMMA GEMMs.
// ---------------------------------------------------------------------------
__global__ __launch_bounds__(128) void k4_merge_out(
    const float* __restrict__ ybuf, const float* __restrict__ mg,
    const float* __restrict__ mbias, const float* __restrict__ mw,
    const float* __restrict__ ow, const float* __restrict__ ob,
    float* __restrict__ out) {
  __shared__ float Yl[4][TILE][4 * D_INNERc];
  __shared__ float Ml[4][TILE][D_INNERc];
  const int wid = threadIdx.y, lane = threadIdx.x;
  const long base = ((long)blockIdx.x * 4 + wid) * TILE;
  float* Ys = &Yl[wid][0][0];
  float* Ms = &Ml[wid][0][0];

  {  // stage 16 pixels x 4 dirs x 32ch into LDS as b128 traffic
    float4* dst = (float4*)Ys;
    for (int i = lane; i < TILE * 32; i += 32) {  // 512 float4 = 2048 floats
      const int m = i >> 5, q = i & 31;           // q: float4 index in row
      const int dir = q >> 3;
      const float4* src = (const float4*)(ybuf + (long)dir * NPIX * D_INNERc +
                                          (base + m) * D_INNERc + (q & 7) * 4);
      dst[i] = *src;
    }
  }
  __syncthreads();

  if (lane < TILE) {                    // LayerNorm over the 128 concat channels
    float s = 0.f;
    for (int c = 0; c < 128; ++c) s += Ys[lane * 128 + c];
    const float mean = s * (1.f / 128.f);
    float v = 0.f;
    for (int c = 0; c < 128; ++c) {
      const float dd = Ys[lane * 128 + c] - mean;
      v += dd * dd;
    }
    const float inv = rsqrtf(v * (1.f / 128.f) + EPSc);
    for (int c = 0; c < 128; ++c)
      Ys[lane * 128 + c] = (Ys[lane * 128 + c] - mean) * inv * mg[c] + mbias[c];
  }
  __syncthreads();

  const int row = lane & 15, hi = lane >> 4;
  for (int nt = 0; nt < 2; ++nt) {      // merge: K=128 -> 32 wmma k-steps
    v8f c = wmma_tile_f32(Ys, 128, mw, 128, 128, nt * 16, D_INNERc - 1, lane);
    const int o = nt * 16 + row;
    for (int r = 0; r < 8; ++r) Ms[(r + hi * 8) * D_INNERc + o] = c[r];
  }
  __syncthreads();
  for (int nt = 0; nt < 2; ++nt) {      // out: K=32 -> 8 wmma k-steps, 20 outs
    v8f c = wmma_tile_f32(Ms, D_INNERc, ow, D_INNERc, D_INNERc,
                          nt * 16, OUT_CHc - 1, lane);
    const int o = nt * 16 + row;
    if (o < OUT_CHc) {
      const float bias = ob[o];
      for (int r = 0; r < 8; ++r) {
        const long p = base + r + hi * 8;
        const long b = p / (long)(Hn * Wn);
        const long rem = p - b * (long)(Hn * Wn);
        out[(b * OUT_CHc + o) * (long)(Hn * Wn) + rem] = c[r] + bias;
      }
    }
  }
}

// ---------------------------------------------------------------------------
extern "C" void kernel_launch(void* const* d_in, const int* in_sizes, int n_in,
                              void* d_out, int out_size, void* d_ws,
                              size_t ws_size, hipStream_t stream) {
  const float* x    = (const float*)d_in[0];
  const float* ng   = (const float*)d_in[1];
  const float* nb   = (const float*)d_in[2];
  const float* ipw  = (const float*)d_in[3];
  const float* dww  = (const float*)d_in[4];
  const float* dwb  = (const float*)d_in[5];
  const float* xw   = (const float*)d_in[6];
  const float* dtw  = (const float*)d_in[7];
  const float* dtb  = (const float*)d_in[8];
  const float* Alog = (const float*)d_in[9];
  const float* Dp   = (const float*)d_in[10];
  const float* mg   = (const float*)d_in[11];
  const float* mb   = (const float*)d_in[12];
  const float* mw   = (const float*)d_in[13];
  const float* ow   = (const float*)d_in[14];
  const float* ob   = (const float*)d_in[15];

  float* ws  = (float*)d_ws;
  float* pre = ws;                         // NPIX*32 (reused as dt_buf)
  float* hb  = ws + (long)NPIX * 32;       // NPIX*32
  float* bc  = ws + (long)NPIX * 64;       // NPIX*16
  float* yb  = ws + (long)NPIX * 80;       // 4 * NPIX*32

  dim3 blk(32, 8);
  k1_ln_inproj<<<NPIX / (16 * 8), blk, 0, stream>>>(x, ng, nb, ipw, pre);
  k2_dwconv<<<(NPIX * D_INNERc) / 256, 256, 0, stream>>>(pre, dww, dwb, hb);

  float* dtB = pre;  // "pre" is dead after k2; reuse as dt buffer
  for (int dir = 0; dir < 4; ++dir) {
    k3a_proj<<<NPIX / (16 * 8), blk, 0, stream>>>(
        hb, xw + dir * (DT_RANKc + 2 * D_STATEc) * D_INNERc,
        dtw + dir * D_INNERc * DT_RANKc, dtb + dir * D_INNERc, dtB, bc);
    k3c_scan<<<(Bn * Hn) / 8, blk, 0, stream>>>(
        hb, dtB, bc, Alog + dir * D_INNERc * D_STATEc, Dp + dir * D_INNERc,
        yb + (long)dir * NPIX * D_INNERc, (dir >= 2) ? 1 : 0, dir & 1);
  }

  dim3 blk4(32, 4);
  k4_merge_out<<<NPIX / (16 * 4), blk4, 0, stream>>>(yb, mg, mb, mw, ow, ob,
                                                     (float*)d_out);
}